// LengthRegulator_7997229105428
// MI455X (gfx1250) — compile-verified
//
#include <hip/hip_runtime.h>
#include <stdint.h>

#define N_B     32
#define L_IN    256
#define D_DIM   512
#define MAXLEN  2048
#define ROW_BYTES (D_DIM * 4)                 /* 2048 B per row */
#define OUT_MAIN  ((size_t)N_B * MAXLEN * D_DIM)

typedef float v4f __attribute__((ext_vector_type(4)));

// ---------------------------------------------------------------------------
// Kernel A: per-batch inclusive scan of durations, build gather index map,
// emit mel_len (as float) into the output tail.
// One block per batch, 256 threads (== L_IN).
// ---------------------------------------------------------------------------
__global__ __launch_bounds__(256) void lr_index_kernel(
    const int* __restrict__ duration,  // (N_B, L_IN)
    int* __restrict__ idxmap,          // (N_B, MAXLEN), -1 => zero row
    float* __restrict__ mel_out)       // (N_B,)
{
  __shared__ int s[L_IN];
  const int n   = blockIdx.x;
  const int tid = threadIdx.x;

  s[tid] = duration[n * L_IN + tid];

  // Hillis-Steele inclusive scan over 256 elements.
  for (int off = 1; off < L_IN; off <<= 1) {
    __syncthreads();
    int addend = (tid >= off) ? s[tid - off] : 0;
    __syncthreads();
    s[tid] += addend;
  }
  __syncthreads();

  const int total = s[L_IN - 1];

  // idx[t] = #{l : csum[l] <= t}  (upper bound), clipped to L_IN-1; -1 if t >= total.
  for (int t = tid; t < MAXLEN; t += 256) {
    int res = -1;
    if (t < total) {
      int pos = 0;
#pragma unroll
      for (int step = 128; step >= 1; step >>= 1) {
        int np = pos + step;
        if (np <= L_IN && s[np - 1] <= t) pos = np;
      }
      res = (pos < (L_IN - 1)) ? pos : (L_IN - 1);
    }
    idxmap[n * MAXLEN + t] = res;
  }

  if (tid == 0) mel_out[n] = (float)total;
}

// ---------------------------------------------------------------------------
// Kernel B: row gather via CDNA5 async global<->LDS DMA path.
// 8 waves per block; each wave owns 4 LDS row slots (4 * 2048 B) and copies
// 4 output rows in a single load->drain->store phase:
//   16 x global_load_async_to_lds_b128  (lane*16B base, 512B inst offsets)
//   s_wait_asynccnt 0
//   16 x global_store_async_from_lds_b128 th:TH_STORE_NT  (write-once stream)
// Final drain is S_ENDPGM's implicit S_WAIT_IDLE.
// ---------------------------------------------------------------------------
__global__ __launch_bounds__(256) void lr_gather_kernel(
    const float* __restrict__ x,      // (N_B, L_IN, D_DIM)
    const int* __restrict__ idxmap,   // (N_B, MAXLEN)
    float* __restrict__ out)          // (N_B, MAXLEN, D_DIM)
{
  __shared__ float smem[8 * 4 * D_DIM];  // 8 waves * 4 slots * 2048 B = 64 KiB

  const int wave = threadIdx.x >> 5;
  const int lane = threadIdx.x & 31;
  const int gw   = blockIdx.x * 8 + wave;

  // Workgroup-relative LDS byte address of this wave's slot region + this
  // lane's 16-byte chunk (low 32 bits of a flat shared address == LDS offset).
  const uint32_t lds_wave =
      (uint32_t)(uintptr_t)(&smem[0]) +
      (uint32_t)(wave * 4 * ROW_BYTES) + (uint32_t)(lane * 16);

  const int rbase = gw * 4;
  int src[4];

  // ---- issue async loads: gathered x rows -> LDS slots ---------------------
#pragma unroll
  for (int sIt = 0; sIt < 4; ++sIt) {
    const int row = rbase + sIt;
    const int idx = idxmap[row];        // same value across the wave
    src[sIt] = idx;
    if (idx >= 0) {
      const int n = row >> 11;          // MAXLEN == 2048 rows per batch
      const void* g =
          (const void*)(x + ((size_t)(n * L_IN + idx) * D_DIM) + lane * 4);
      const uint32_t l = lds_wave + (uint32_t)(sIt * ROW_BYTES);
      asm volatile(
          "global_load_async_to_lds_b128 %0, %1, off\n\t"
          "global_load_async_to_lds_b128 %0, %1, off offset:512\n\t"
          "global_load_async_to_lds_b128 %0, %1, off offset:1024\n\t"
          "global_load_async_to_lds_b128 %0, %1, off offset:1536"
          :: "v"(l), "v"(g) : "memory");
    }
  }

  // loads must land in LDS before the stores read it
  asm volatile("s_wait_asynccnt 0x0" ::: "memory");

  // ---- issue async stores: LDS slots -> contiguous output rows -------------
#pragma unroll
  for (int sIt = 0; sIt < 4; ++sIt) {
    const int row = rbase + sIt;
    float* o = out + (size_t)row * D_DIM + lane * 4;
    if (src[sIt] >= 0) {
      const uint32_t l = lds_wave + (uint32_t)(sIt * ROW_BYTES);
      asm volatile(
          "global_store_async_from_lds_b128 %1, %0, off th:TH_STORE_NT\n\t"
          "global_store_async_from_lds_b128 %1, %0, off offset:512 th:TH_STORE_NT\n\t"
          "global_store_async_from_lds_b128 %1, %0, off offset:1024 th:TH_STORE_NT\n\t"
          "global_store_async_from_lds_b128 %1, %0, off offset:1536 th:TH_STORE_NT"
          :: "v"(l), "v"((void*)o) : "memory");
    } else {
      // zero-fill row; write-once data -> non-temporal stores
      v4f z = {0.f, 0.f, 0.f, 0.f};
      __builtin_nontemporal_store(z, (v4f*)(o));
      __builtin_nontemporal_store(z, (v4f*)(o + 128));
      __builtin_nontemporal_store(z, (v4f*)(o + 256));
      __builtin_nontemporal_store(z, (v4f*)(o + 384));
    }
  }
  // implicit S_WAIT_IDLE at S_ENDPGM drains outstanding async stores
}

// ---------------------------------------------------------------------------
extern "C" void kernel_launch(void* const* d_in, const int* in_sizes, int n_in,
                              void* d_out, int out_size, void* d_ws, size_t ws_size,
                              hipStream_t stream) {
  const float* x        = (const float*)d_in[0];
  const int*   duration = (const int*)d_in[1];   // integer input -> const int*
  // d_in[2] = max_len scalar (compile-time 2048 here)

  float* out    = (float*)d_out;
  int*   idxmap = (int*)d_ws;                    // N_B * MAXLEN ints = 256 KiB

  lr_index_kernel<<<N_B, 256, 0, stream>>>(duration, idxmap, out + OUT_MAIN);

  // rows = N_B*MAXLEN = 65536; 4 rows per wave, 8 waves per block -> 2048 blocks
  lr_gather_kernel<<<(N_B * MAXLEN) / 32, 256, 0, stream>>>(x, idxmap, out);
}